// GNN_75076028334671
// MI455X (gfx1250) — compile-verified
//
#include <hip/hip_runtime.h>

// 3-layer GCN (PyG GCNConv, symmetric norm + self loops) for MI455X / gfx1250.
//
// Restructure: norm[e] = dinv[row]*dinv[col] is never materialized.
//   hs   = dinv ⊙ (h @ W)                      (WMMA f32 GEMM epilogue)
//   agg  = hs (self-loop init, same epilogue)
//   agg[col] += hs[row]   per edge             (f32 global atomics, L2-resident)
//   h'   = relu(dinv ⊙ agg + b)                (finish pass)
// All GEMMs use V_WMMA_F32_16X16X4_F32 (exact f32, H=15 padded to 16).
// N = 100000 is divisible by 16 -> no tail tiles, EXEC all-ones for WMMA.

typedef __attribute__((ext_vector_type(2))) float v2f;
typedef __attribute__((ext_vector_type(8))) float v8f;

#define DFEAT 128
#define HFEAT 15
#define HPAD  16

__device__ __forceinline__ void atomicAddF(float* p, float v) {
  unsafeAtomicAdd(p, v);  // lowers to global_atomic_add_f32 (no CAS loop)
}

// ---------------- degree / rsqrt norm ----------------
__global__ void k_deg_init(float* __restrict__ deg, int n) {
  int i = blockIdx.x * blockDim.x + threadIdx.x;
  if (i < n) deg[i] = 1.0f;  // self-loop contribution
}

__global__ void k_deg_count(const int* __restrict__ col, float* __restrict__ deg, int e) {
  int i = blockIdx.x * blockDim.x + threadIdx.x;
  if (i < e) atomicAddF(&deg[col[i]], 1.0f);
}

__global__ void k_deg_rsqrt(float* __restrict__ deg, int n) {
  int i = blockIdx.x * blockDim.x + threadIdx.x;
  if (i < n) deg[i] = rsqrtf(deg[i]);  // deg >= 1 always (self loops)
}

// ---------------- GEMM (N x K) @ (K x 15 padded to 16), scaled by dinv ----------------
// One wave computes one 16x16 output tile with K/4 V_WMMA_F32_16X16X4_F32 ops.
// A fragment layout (16x4 f32): lane = M + 16*(K>>1), vgpr = K&1.
// Epilogue writes hs = c*dinv[node] and initializes agg = hs (self loop).
template <int K>
__global__ void k_gemm_scale(const float* __restrict__ A,   // N x K row-major
                             const float* __restrict__ W,   // wrows x 15 row-major
                             int wrows,
                             const float* __restrict__ dinv,
                             float* __restrict__ hs,
                             float* __restrict__ agg,
                             int numTiles) {
  __shared__ float wlds[K][HPAD];  // zero-padded weights
  for (int i = threadIdx.x; i < K * HPAD; i += blockDim.x) {
    int k = i >> 4, n = i & 15;
    wlds[k][n] = (k < wrows && n < HFEAT) ? W[k * HFEAT + n] : 0.0f;
  }
  __syncthreads();

  int wave = threadIdx.x >> 5;
  int lane = threadIdx.x & 31;
  int tile = blockIdx.x * (blockDim.x >> 5) + wave;
  if (tile >= numTiles) return;  // whole-wave uniform: EXEC stays all-ones

  int m  = lane & 15;          // A row within tile; also B/D column
  int kh = (lane >> 4) << 1;   // k sub-offset for this half-wave
  const float* arow = A + (size_t)(tile * 16 + m) * K;

  v8f c = {};
#pragma unroll
  for (int k0 = 0; k0 < K; k0 += 4) {
    v2f a, b;
    a.x = arow[k0 + kh];
    a.y = arow[k0 + kh + 1];
    b.x = wlds[k0 + kh][m];
    b.y = wlds[k0 + kh + 1][m];
    c = __builtin_amdgcn_wmma_f32_16x16x4_f32(false, a, false, b, (short)0, c,
                                              false, false);
  }

  // D layout: vgpr i -> row (i + 8*(lane>=16)), col = lane&15
  int mbase = (lane >> 4) << 3;
#pragma unroll
  for (int i = 0; i < 8; ++i) {
    int node = tile * 16 + mbase + i;
    float v = c[i] * dinv[node];
    hs[node * HPAD + m]  = v;  // scaled features for gather
    agg[node * HPAD + m] = v;  // self-loop term initializes the segment sum
  }
}

// ---------------- edge scatter: agg[col] += hs[row] ----------------
// 4 threads per edge, float4 gather (hs row is 64B), 4 f32 atomics each.
// hs[...][15] is exactly 0 (padded weight column) so the extra lane is harmless.
__global__ void k_scatter(const int* __restrict__ row, const int* __restrict__ col,
                          const float* __restrict__ hs, float* __restrict__ agg,
                          int e) {
  int t = blockIdx.x * blockDim.x + threadIdx.x;
  int eid = t >> 2;
  if (eid >= e) return;
  int q = (t & 3) << 2;
  int r  = row[eid];
  int cc = col[eid];
  const float4 v = *reinterpret_cast<const float4*>(hs + (size_t)r * HPAD + q);
  float* dst = agg + (size_t)cc * HPAD + q;
  atomicAddF(dst + 0, v.x);
  atomicAddF(dst + 1, v.y);
  atomicAddF(dst + 2, v.z);
  atomicAddF(dst + 3, v.w);
}

// ---------------- finish: h = relu(dinv*agg + b), keep pad column 0 ----------------
__global__ void k_finish(const float* __restrict__ agg, const float* __restrict__ dinv,
                         const float* __restrict__ b, float* __restrict__ hout,
                         int n) {
  int t = blockIdx.x * blockDim.x + threadIdx.x;
  if (t >= n * HPAD) return;
  int v = t >> 4, nn = t & 15;
  float x = 0.0f;
  if (nn < HFEAT) x = fmaxf(dinv[v] * agg[t] + b[nn], 0.0f);
  hout[t] = x;
}

// ---------------- output GEMM: (N x 16) @ (15 x 128 padded to 16) + bc ----------------
__global__ void k_gemm_out(const float* __restrict__ A,   // N x 16 (h3)
                           const float* __restrict__ Wc,  // 15 x 128
                           const float* __restrict__ bc,  // 128
                           float* __restrict__ out,       // N x 128
                           int numNodeTiles) {
  __shared__ float wlds[HPAD][DFEAT];  // 8KB, row 15 zero-padded
  for (int i = threadIdx.x; i < HPAD * DFEAT; i += blockDim.x) {
    int k = i >> 7, n = i & 127;
    wlds[k][n] = (k < HFEAT) ? Wc[k * DFEAT + n] : 0.0f;
  }
  __syncthreads();

  int wave  = threadIdx.x >> 5;
  int lane  = threadIdx.x & 31;
  int wtile = blockIdx.x * (blockDim.x >> 5) + wave;
  if (wtile >= numNodeTiles * 8) return;
  int tile = wtile >> 3;
  int n0   = (wtile & 7) << 4;

  int m  = lane & 15;
  int kh = (lane >> 4) << 1;
  const float* arow = A + (size_t)(tile * 16 + m) * HPAD;

  v8f c = {};
#pragma unroll
  for (int k0 = 0; k0 < HPAD; k0 += 4) {
    v2f a, b;
    a.x = arow[k0 + kh];
    a.y = arow[k0 + kh + 1];
    b.x = wlds[k0 + kh][n0 + m];
    b.y = wlds[k0 + kh + 1][n0 + m];
    c = __builtin_amdgcn_wmma_f32_16x16x4_f32(false, a, false, b, (short)0, c,
                                              false, false);
  }

  float bias = bc[n0 + m];
  int mbase = (lane >> 4) << 3;
#pragma unroll
  for (int i = 0; i < 8; ++i) {
    int node = tile * 16 + mbase + i;
    out[(size_t)node * DFEAT + n0 + m] = c[i] + bias;
  }
}

extern "C" void kernel_launch(void* const* d_in, const int* in_sizes, int n_in,
                              void* d_out, int out_size, void* d_ws, size_t ws_size,
                              hipStream_t stream) {
  const float* x  = (const float*)d_in[0];
  const int*   ei = (const int*)d_in[1];  // harness contract: integer -> const int*
  const float* W1 = (const float*)d_in[2];
  const float* b1 = (const float*)d_in[3];
  const float* W2 = (const float*)d_in[4];
  const float* b2 = (const float*)d_in[5];
  const float* W3 = (const float*)d_in[6];
  const float* b3 = (const float*)d_in[7];
  const float* Wc = (const float*)d_in[8];
  const float* bc = (const float*)d_in[9];

  const int N = in_sizes[0] / DFEAT;  // 100000 (divisible by 16)
  const int E = in_sizes[1] / 2;      // 6400000
  const int* row = ei;      // edge_index[0] : source (gather)
  const int* col = ei + E;  // edge_index[1] : target (segment sum)

  // workspace layout: dinv | hs | agg | h   (~19.6 MB)
  float* dinv = (float*)d_ws;
  float* B0 = dinv + N;                 // hs
  float* B1 = B0 + (size_t)N * HPAD;    // agg
  float* B2 = B1 + (size_t)N * HPAD;    // h (layer input)
  float* outf = (float*)d_out;

  const int T = 256;                    // 8 waves/block (wave32)
  const int numTiles = N / 16;          // 6250
  dim3 gN((N + T - 1) / T);
  dim3 gE((E + T - 1) / T);
  dim3 gE4((E * 4 + T - 1) / T);
  dim3 gF((N * HPAD + T - 1) / T);
  dim3 gG((numTiles + 7) / 8);
  dim3 gO((numTiles * 8 + 7) / 8);

  // ---- degrees -> dinv = rsqrt(deg) ----
  k_deg_init <<<gN, T, 0, stream>>>(dinv, N);
  k_deg_count<<<gE, T, 0, stream>>>(col, dinv, E);
  k_deg_rsqrt<<<gN, T, 0, stream>>>(dinv, N);

  // ---- layer 1: x (N x 128) ----
  k_gemm_scale<DFEAT><<<gG, T, 0, stream>>>(x, W1, DFEAT, dinv, B0, B1, numTiles);
  k_scatter<<<gE4, T, 0, stream>>>(row, col, B0, B1, E);
  k_finish <<<gF,  T, 0, stream>>>(B1, dinv, b1, B2, N);

  // ---- layer 2 ----
  k_gemm_scale<HPAD><<<gG, T, 0, stream>>>(B2, W2, HFEAT, dinv, B0, B1, numTiles);
  k_scatter<<<gE4, T, 0, stream>>>(row, col, B0, B1, E);
  k_finish <<<gF,  T, 0, stream>>>(B1, dinv, b2, B2, N);

  // ---- layer 3 ----
  k_gemm_scale<HPAD><<<gG, T, 0, stream>>>(B2, W3, HFEAT, dinv, B0, B1, numTiles);
  k_scatter<<<gE4, T, 0, stream>>>(row, col, B0, B1, E);
  k_finish <<<gF,  T, 0, stream>>>(B1, dinv, b3, B2, N);

  // ---- output projection ----
  k_gemm_out<<<gO, T, 0, stream>>>(B2, Wc, bc, outf, numTiles);
}